// RagInfoNceLoss_2886218023667
// MI455X (gfx1250) — compile-verified
//
#include <hip/hip_runtime.h>
#include <math.h>

// Problem constants (fixed by the reference's setup_inputs)
#define NPIX  (1024*1024)   // H*W pixels
#define NCH   32            // channels
#define NSEG  1024          // segments
#define NEDGE 8192          // RAG edges
#define EPSF  1e-8f         // torch cosine_similarity eps
#define GRID_ACC 128        // blocks for segment accumulation (128*8192 = NPIX)
#define PPB   (NPIX / GRID_ACC)
// TAU = 0.5  ->  /TAU == *2.0f

typedef __attribute__((ext_vector_type(2))) float v2f;
typedef __attribute__((ext_vector_type(8))) float v8f;

// ---------------------------------------------------------------------------
// K0: zero segment sums+counts and the 3 double accumulators
// ---------------------------------------------------------------------------
__global__ __launch_bounds__(256) void zero_ws_kernel(float* __restrict__ sc,
                                                      double* __restrict__ acc) {
  int i = blockIdx.x * blockDim.x + threadIdx.x;
  if (i < NSEG * NCH + NSEG) sc[i] = 0.0f;
  if (i < 3) acc[i] = 0.0;
}

// ---------------------------------------------------------------------------
// K1: segment sums + counts.
// CDNA5: 320KB LDS per WGP -> the ENTIRE accumulator set (1024x32 sums +
// 1024 counts = 132KB) fits in one workgroup's LDS. Accumulate per-block
// partials with ds_add_f32, flush once per block -> global f32 atomics drop
// from 34.6M to ~4.3M while the hot adds run at per-WGP LDS throughput.
// emb layout is channel-major: emb[c*NPIX + n] (coalesced across lanes).
// ---------------------------------------------------------------------------
__global__ __launch_bounds__(256) void seg_accum_kernel(const float* __restrict__ emb,
                                                        const int* __restrict__ seg,
                                                        float* __restrict__ sums,
                                                        float* __restrict__ counts) {
  __shared__ float lsum[NSEG * NCH];   // 128KB
  __shared__ float lcnt[NSEG];         // 4KB

  for (int i = threadIdx.x; i < NSEG * NCH; i += 256) lsum[i] = 0.0f;
  for (int i = threadIdx.x; i < NSEG; i += 256)       lcnt[i] = 0.0f;
  __syncthreads();

  const int base = blockIdx.x * PPB;
  for (int t = threadIdx.x; t < PPB; t += 256) {
    const int n = base + t;
    const int s = seg[n];
    atomicAdd(&lcnt[s], 1.0f);
#pragma unroll
    for (int c = 0; c < NCH; ++c)
      atomicAdd(&lsum[s * NCH + c], emb[(size_t)c * NPIX + n]);
  }
  __syncthreads();

  for (int i = threadIdx.x; i < NSEG * NCH; i += 256) atomicAdd(&sums[i], lsum[i]);
  for (int i = threadIdx.x; i < NSEG; i += 256)       atomicAdd(&counts[i], lcnt[i]);
}

// ---------------------------------------------------------------------------
// K2: means = sums / max(counts,1); rnorm = 1/max(||mean||, EPS)
// ---------------------------------------------------------------------------
__global__ __launch_bounds__(256) void means_kernel(const float* __restrict__ sums,
                                                    const float* __restrict__ counts,
                                                    float* __restrict__ means,
                                                    float* __restrict__ rnorm) {
  int s = blockIdx.x * blockDim.x + threadIdx.x;
  if (s >= NSEG) return;
  float inv = 1.0f / fmaxf(counts[s], 1.0f);
  float ssq = 0.0f;
#pragma unroll
  for (int c = 0; c < NCH; ++c) {
    float m = sums[s * NCH + c] * inv;
    means[s * NCH + c] = m;
    ssq = fmaf(m, m, ssq);
  }
  rnorm[s] = 1.0f / fmaxf(sqrtf(ssq), EPSF);
}

// ---------------------------------------------------------------------------
// K3: inter edges -> acc[1] += sum(exp(cos/tau)). Grid is exactly NEDGE threads.
// ---------------------------------------------------------------------------
__global__ __launch_bounds__(256) void inter_kernel(const int* __restrict__ edges,
                                                    const float* __restrict__ means,
                                                    const float* __restrict__ rnorm,
                                                    double* __restrict__ acc) {
  int e = blockIdx.x * blockDim.x + threadIdx.x;
  int a = edges[e];
  int b = edges[NEDGE + e];
  float dot = 0.0f;
#pragma unroll
  for (int c = 0; c < NCH; ++c)
    dot = fmaf(means[a * NCH + c], means[b * NCH + c], dot);
  float v  = dot * rnorm[a] * rnorm[b] * 2.0f;   // / TAU
  float ex = expf(v);
#pragma unroll
  for (int off = 16; off > 0; off >>= 1)
    ex += __shfl_xor(ex, off, 32);
  if ((threadIdx.x & 31) == 0)
    atomicAdd(&acc[1], (double)ex);
}

// ---------------------------------------------------------------------------
// K4: intra via V_WMMA_F32_16X16X4_F32. One wave = one 16-pixel tile.
//   A (16x4 f32, chained over 8 K-chunks) = pixel embeddings
//   B (4x16 f32)                          = gathered segment means
//   diag(16x16 D) = per-pixel dot(mean[seg], emb)
// A-fragment layout (ISA 7.12.2, f32 16x4): lanes 0-15 rows M=0..15,
//   VGPR0 = K=0 (K=2 for lanes>=16), VGPR1 = K=1 (K=3).  B assumed symmetric.
// D layout: M = vgpr + 8*(lane>=16), N = lane&15 -> diag p<8 at (lane p, c[p]),
//   diag p>=8 at (lane p+16, c[p-8]).
// ---------------------------------------------------------------------------
__global__ __launch_bounds__(256) void intra_wmma_kernel(const float* __restrict__ emb,
                                                         const int* __restrict__ seg,
                                                         const float* __restrict__ means,
                                                         const float* __restrict__ rnorm,
                                                         double* __restrict__ acc) {
  const int lane = threadIdx.x & 31;
  const int wave = threadIdx.x >> 5;
  const int tile = blockIdx.x * 8 + wave;   // grid sized so every tile is full
  const int n0   = tile * 16;

  const int rc = lane & 15;         // A row (= pixel within tile) and B column
  const int kh = (lane >> 4) * 2;   // K sub-offset within a 4-chunk: 0 or 2

  const int   n      = n0 + rc;
  const int   s      = seg[n];      // segment of this lane's pixel/column
  const float rn_seg = rnorm[s];
  __builtin_prefetch(&means[s * NCH], 0, 0);   // global_prefetch_b8 on the gather

  v8f   cacc = {0.f, 0.f, 0.f, 0.f, 0.f, 0.f, 0.f, 0.f};
  float ssq  = 0.0f;

#pragma unroll
  for (int k0 = 0; k0 < 8; ++k0) {
    const int ka = 4 * k0 + kh;
    float ax = emb[(size_t)(ka + 0) * NPIX + n];
    float ay = emb[(size_t)(ka + 1) * NPIX + n];
    ssq = fmaf(ax, ax, ssq);
    ssq = fmaf(ay, ay, ssq);
    v2f A; A.x = ax; A.y = ay;
    v2f B; B.x = means[s * NCH + ka + 0]; B.y = means[s * NCH + ka + 1];
    cacc = __builtin_amdgcn_wmma_f32_16x16x4_f32(false, A, false, B,
                                                 (short)0, cacc, false, false);
  }

  // lanes L and L^16 hold complementary channels of the same pixel row
  ssq += __shfl_xor(ssq, 16, 32);

  // diagonal extraction per documented C/D layout
  const bool active = (lane < 8) || (lane >= 24);
  const int  idx    = (lane < 8) ? lane : (lane - 24);
  float dv = cacc[0];
  dv = (idx == 1) ? cacc[1] : dv;
  dv = (idx == 2) ? cacc[2] : dv;
  dv = (idx == 3) ? cacc[3] : dv;
  dv = (idx == 4) ? cacc[4] : dv;
  dv = (idx == 5) ? cacc[5] : dv;
  dv = (idx == 6) ? cacc[6] : dv;
  dv = (idx == 7) ? cacc[7] : dv;

  float intra = dv * rn_seg * (1.0f / fmaxf(sqrtf(ssq), EPSF)) * 2.0f; // / TAU
  float vexp  = active ? expf(intra) : 0.0f;
  float vint  = active ? intra       : 0.0f;
#pragma unroll
  for (int off = 16; off > 0; off >>= 1) {
    vexp += __shfl_xor(vexp, off, 32);
    vint += __shfl_xor(vint, off, 32);
  }
  if (lane == 0) {
    atomicAdd(&acc[0], (double)vexp);   // sum(exp(intra))
    atomicAdd(&acc[2], (double)vint);   // sum(intra)
  }
}

// ---------------------------------------------------------------------------
// K5: loss = log(denom) - mean(intra)
// ---------------------------------------------------------------------------
__global__ void finalize_kernel(const double* __restrict__ acc, float* __restrict__ out) {
  if (threadIdx.x == 0 && blockIdx.x == 0) {
    double denom = acc[0] + acc[1];
    out[0] = (float)(log(denom) - acc[2] / (double)NPIX);
  }
}

// ---------------------------------------------------------------------------
extern "C" void kernel_launch(void* const* d_in, const int* in_sizes, int n_in,
                              void* d_out, int out_size, void* d_ws, size_t ws_size,
                              hipStream_t stream) {
  const float* emb   = (const float*)d_in[0];  // (1,32,1024,1024) f32, channel-major
  const int*   seg   = (const int*)d_in[1];    // (1,1,1024,1024) i32
  const int*   edges = (const int*)d_in[2];    // (2,8192) i32
  float*       out   = (float*)d_out;          // scalar f32

  // workspace layout
  char*   w      = (char*)d_ws;
  float*  sums   = (float*)w;                          // 32768 f
  float*  counts = sums + NSEG * NCH;                  // 1024 f (contiguous after sums)
  float*  means  = counts + NSEG;                      // 32768 f
  float*  rnorm  = means + NSEG * NCH;                 // 1024 f
  double* acc    = (double*)(w + (size_t)(2 * NSEG * NCH + 2 * NSEG) * sizeof(float)); // 8B aligned

  zero_ws_kernel<<<(NSEG * NCH + NSEG + 255) / 256, 256, 0, stream>>>(sums, acc);
  seg_accum_kernel<<<GRID_ACC, 256, 0, stream>>>(emb, seg, sums, counts);
  means_kernel<<<(NSEG + 255) / 256, 256, 0, stream>>>(sums, counts, means, rnorm);
  inter_kernel<<<NEDGE / 256, 256, 0, stream>>>(edges, means, rnorm, acc);
  intra_wmma_kernel<<<NPIX / (16 * 8), 256, 0, stream>>>(emb, seg, means, rnorm, acc);
  finalize_kernel<<<1, 1, 0, stream>>>(acc, out);
}